// MPNNModel_17858474017314
// MI455X (gfx1250) — compile-verified
//
#include <hip/hip_runtime.h>
#include <hip/hip_bf16.h>
#include <math.h>

typedef __attribute__((ext_vector_type(2))) float v2f;
typedef __attribute__((ext_vector_type(8))) float v8f;

#define HID 64
#define NGRAPH 256
#define BN_EPS 1e-5f

// ---------------------------------------------------------------- utilities
__global__ void fill_f32(float* __restrict__ p, float v, long long n) {
    long long i = (long long)blockIdx.x * blockDim.x + threadIdx.x;
    if (i < n) p[i] = v;
}

// deg[i] starts at 2.0 (two explicit self loops); add 1 per incoming edge.
__global__ void deg_accum(const int* __restrict__ col, float* __restrict__ deg, int E) {
    int e = blockIdx.x * blockDim.x + threadIdx.x;
    if (e < E) atomicAdd(&deg[col[e]], 1.0f);
}

__global__ void deg_to_dinv(float* __restrict__ d, int n) {
    int i = blockIdx.x * blockDim.x + threadIdx.x;
    if (i < n) d[i] = rsqrtf(d[i]);   // deg >= 2 always
}

// ---------------------------------------------------------------- WMMA GEMM
// T[n x 64] = H[n x K] @ W[K x 64] via V_WMMA_F32_16X16X4_F32 (fp32, full precision).
// One wave -> one 16x16 D tile; 8 waves/block: 2 row-tiles x 4 col-tiles (32 rows).
// Epilogue also seeds the aggregation buffer with the analytic self-loop + bias term:
//   O[m][c] = 2*dinv[m]^2 * T[m][c] + bias[c]
// so edge atomics accumulate on top and the finish pass is just ReLU.
template <int K>
__global__ void gemm_wmma_f32(const float* __restrict__ H, const float* __restrict__ W,
                              const float* __restrict__ dinv, const float* __restrict__ bias,
                              float* __restrict__ T, float* __restrict__ O, int n) {
    const int lane    = threadIdx.x & 31;
    const int wave    = threadIdx.x >> 5;                 // 0..7
    const int rowBase = (blockIdx.x * 2 + (wave >> 2)) * 16;
    const int colBase = (wave & 3) * 16;
    if (rowBase >= n) return;                             // wave-uniform

    const int half = lane >> 4;      // 0: lanes 0-15 (K=0,1) ; 1: lanes 16-31 (K=2,3)
    const int lid  = lane & 15;
    // Clamp instead of predicate: keeps EXEC all-ones through the WMMA loop
    // (out-of-range rows are dropped at the D write).
    const int aRow = min(rowBase + lid, n - 1);
    const float* __restrict__ hA = H + (size_t)aRow * K + half * 2;
    const float* __restrict__ wB = W + (size_t)(half * 2) * 64 + colBase + lid;

    v8f c = {};
#pragma unroll
    for (int k = 0; k < K; k += 4) {
        v2f a; a.x = hA[k];          a.y = hA[k + 1];
        v2f b; b.x = wB[(size_t)k * 64]; b.y = wB[(size_t)(k + 1) * 64];
        c = __builtin_amdgcn_wmma_f32_16x16x4_f32(
                false, a, false, b, (short)0, c, false, false);
    }
    const float bc = bias[colBase + lid];
    // D layout: VGPR r -> lanes 0-15: M=r, N=lid ; lanes 16-31: M=r+8, N=lid
#pragma unroll
    for (int r = 0; r < 8; ++r) {
        const int m = rowBase + r + half * 8;
        if (m < n) {
            const float d  = c[r];
            const float di = dinv[m];
            const size_t o = (size_t)m * 64 + colBase + lid;
            T[o] = d;
            O[o] = 2.0f * di * di * d + bc;
        }
    }
}

// ---------------------------------------------------------------- message pass
// For each edge e: O[col[e]] += T[row[e]] * dinv[row]*dinv[col]
// 16 threads per edge, float4 per thread -> coalesced gather + 4 f32 atomics.
__global__ void edge_agg(const int* __restrict__ row, const int* __restrict__ col,
                         const float* __restrict__ dinv, const float* __restrict__ T,
                         float* __restrict__ O, int E) {
    long long tid = (long long)blockIdx.x * blockDim.x + threadIdx.x;
    if (tid >= (long long)E * 16) return;
    const int e = (int)(tid >> 4);
    const int d = ((int)tid & 15) * 4;
    const int r  = row[e];
    const int cl = col[e];
    const float nrm = dinv[r] * dinv[cl];
    const float4 tv = *(const float4*)(T + (size_t)r * 64 + d);
    float* o = O + (size_t)cl * 64 + d;
    atomicAdd(o + 0, tv.x * nrm);
    atomicAdd(o + 1, tv.y * nrm);
    atomicAdd(o + 2, tv.z * nrm);
    atomicAdd(o + 3, tv.w * nrm);
}

// In-place ReLU (self-loop + bias already folded in by the GEMM epilogue).
__global__ void relu_inplace(float* __restrict__ A, long long n) {
    long long idx = (long long)blockIdx.x * blockDim.x + threadIdx.x;
    if (idx < n) A[idx] = fmaxf(A[idx], 0.0f);
}

// ---------------------------------------------------------------- pooling
__global__ void count_nodes(const int* __restrict__ batch, float* __restrict__ cnts, int n) {
    int i = blockIdx.x * blockDim.x + threadIdx.x;
    if (i < n) atomicAdd(&cnts[batch[i]], 1.0f);
}

// h >= 0 after ReLU -> unsigned bit pattern is order-preserving; init 0 handles empties.
__global__ void pool_kernel(const float* __restrict__ H, const int* __restrict__ batch,
                            float* __restrict__ msum, unsigned int* __restrict__ mmax, int n) {
    long long idx = (long long)blockIdx.x * blockDim.x + threadIdx.x;
    if (idx >= (long long)n * 64) return;
    const int i = (int)(idx >> 6);
    const int d = (int)(idx & 63);
    const int g = batch[i];
    const float h = H[idx];
    atomicAdd(&msum[g * 64 + d], h);
    atomicMax(&mmax[g * 64 + d], __float_as_uint(h));
}

// ---------------------------------------------------------------- MLP head, one block per graph
__global__ void mlp_head(const float* __restrict__ msum, const float* __restrict__ mmax,
                         const float* __restrict__ cnts,
                         const float* fc1W, const float* fc1b,
                         const float* g1, const float* b1, const float* m1, const float* v1,
                         const float* fc2W, const float* fc2b,
                         const float* g2, const float* b2, const float* m2, const float* v2,
                         const float* fc3W, const float* fc3b,
                         float* __restrict__ out) {
    __shared__ float z[2 * HID];
    __shared__ float z1[160];
    __shared__ float z2[90];
    const int g = blockIdx.x;
    const int t = threadIdx.x;
    const float cnt = fmaxf(cnts[g], 1.0f);
    for (int i = t; i < HID; i += blockDim.x) {
        z[i]       = msum[g * HID + i] / cnt;
        z[HID + i] = mmax[g * HID + i];
    }
    __syncthreads();
    if (t < 160) {
        float acc = fc1b[t];
        for (int i = 0; i < 2 * HID; ++i) acc += z[i] * fc1W[i * 160 + t];
        acc = (acc - m1[t]) * rsqrtf(v1[t] + BN_EPS) * g1[t] + b1[t];
        z1[t] = fmaxf(acc, 0.0f);
    }
    __syncthreads();
    if (t < 90) {
        float acc = fc2b[t];
        for (int i = 0; i < 160; ++i) acc += z1[i] * fc2W[i * 90 + t];
        acc = (acc - m2[t]) * rsqrtf(v2[t] + BN_EPS) * g2[t] + b2[t];
        z2[t] = fmaxf(acc, 0.0f);
    }
    __syncthreads();
    if (t == 0) {
        float acc = fc3b[0];
        for (int i = 0; i < 90; ++i) acc += z2[i] * fc3W[i];
        out[g] = 1.0f / (1.0f + expf(-acc));
    }
}

// ---------------------------------------------------------------- launch
static inline size_t align256(size_t x) { return (x + 255) & ~(size_t)255; }
static inline int cdiv(long long a, int b) { return (int)((a + b - 1) / b); }

extern "C" void kernel_launch(void* const* d_in, const int* in_sizes, int n_in,
                              void* d_out, int out_size, void* d_ws, size_t ws_size,
                              hipStream_t stream) {
    const int N = in_sizes[0] / 128;   // nodes
    const int E = in_sizes[1] / 2;     // edges

    const float* x     = (const float*)d_in[0];
    const int*   row   = (const int*)d_in[1];
    const int*   col   = row + E;
    const int*   batch = (const int*)d_in[2];
    const float* Wl[4] = {(const float*)d_in[3], (const float*)d_in[5],
                          (const float*)d_in[7], (const float*)d_in[9]};
    const float* bl[4] = {(const float*)d_in[4], (const float*)d_in[6],
                          (const float*)d_in[8], (const float*)d_in[10]};
    const float* fc1W = (const float*)d_in[11]; const float* fc1b = (const float*)d_in[12];
    const float* g1 = (const float*)d_in[13]; const float* b1 = (const float*)d_in[14];
    const float* m1 = (const float*)d_in[15]; const float* v1 = (const float*)d_in[16];
    const float* fc2W = (const float*)d_in[17]; const float* fc2b = (const float*)d_in[18];
    const float* g2 = (const float*)d_in[19]; const float* b2 = (const float*)d_in[20];
    const float* m2 = (const float*)d_in[21]; const float* v2 = (const float*)d_in[22];
    const float* fc3W = (const float*)d_in[23]; const float* fc3b = (const float*)d_in[24];
    float* out = (float*)d_out;

    // workspace carve-up
    char* ws = (char*)d_ws;
    size_t off = 0;
    float* dinv = (float*)(ws + off); off += align256((size_t)N * 4);
    float* tbuf = (float*)(ws + off); off += align256((size_t)N * 64 * 4);
    float* hA   = (float*)(ws + off); off += align256((size_t)N * 64 * 4);
    float* hB   = (float*)(ws + off); off += align256((size_t)N * 64 * 4);
    float* msum = (float*)(ws + off); off += align256((size_t)NGRAPH * 64 * 4);
    float* mmax = (float*)(ws + off); off += align256((size_t)NGRAPH * 64 * 4);
    float* cnts = (float*)(ws + off); off += align256((size_t)NGRAPH * 4);
    (void)ws_size; (void)n_in; (void)out_size;

    const int TB = 256;
    const long long NF = (long long)N * 64;

    // degrees -> dinv  (init 2.0 for the doubled self loops)
    fill_f32<<<cdiv(N, TB), TB, 0, stream>>>(dinv, 2.0f, N);
    deg_accum<<<cdiv(E, TB), TB, 0, stream>>>(col, dinv, E);
    deg_to_dinv<<<cdiv(N, TB), TB, 0, stream>>>(dinv, N);

    // 4 GCN layers: WMMA GEMM (+ seeded self-loop/bias) -> edge scatter -> ReLU
    const float* hin = x;
    float* houts[4] = {hA, hB, hA, hB};
    const int gemmGrid = cdiv(N, 32);           // 8 waves/block cover 32 rows x 64 cols
    for (int l = 0; l < 4; ++l) {
        float* hout = houts[l];
        if (l == 0)
            gemm_wmma_f32<128><<<gemmGrid, TB, 0, stream>>>(hin, Wl[l], dinv, bl[l],
                                                            tbuf, hout, N);
        else
            gemm_wmma_f32<64><<<gemmGrid, TB, 0, stream>>>(hin, Wl[l], dinv, bl[l],
                                                           tbuf, hout, N);
        edge_agg<<<cdiv((long long)E * 16, TB), TB, 0, stream>>>(row, col, dinv, tbuf, hout, E);
        relu_inplace<<<cdiv(NF, TB), TB, 0, stream>>>(hout, NF);
        hin = hout;
    }

    // pooling buffers (msum+mmax+cnts contiguous; mmax zero-bits == 0.0f)
    fill_f32<<<cdiv((long long)(NGRAPH * 64 * 2 + NGRAPH), TB), TB, 0, stream>>>(
        msum, 0.0f, (long long)NGRAPH * 64 * 2 + NGRAPH);
    count_nodes<<<cdiv(N, TB), TB, 0, stream>>>(batch, cnts, N);
    pool_kernel<<<cdiv(NF, TB), TB, 0, stream>>>((const float*)hin, batch, msum,
                                                 (unsigned int*)mmax, N);

    // MLP head: one block per graph
    mlp_head<<<NGRAPH, 160, 0, stream>>>(msum, mmax, cnts,
                                         fc1W, fc1b, g1, b1, m1, v1,
                                         fc2W, fc2b, g2, b2, m2, v2,
                                         fc3W, fc3b, out);
}